// GlobalTopkExtremaPooling2D_90159953478163
// MI455X (gfx1250) — compile-verified
//
#include <hip/hip_runtime.h>
#include <math.h>

// Problem constants (from reference): inputs [16,128,128,256] f32, K=8.
#define B_    16
#define H_    128
#define W_    128
#define C_    256
#define K_    8
#define HW_   (H_ * W_)
#define S_    64               // segments per (b,c); phase 2: 2 segments per lane
#define ROWS_ (HW_ / S_)       // 256 spatial positions per segment
#define CPT_  4                // channels per thread (float4 stream)
#define TPB1_ (C_ / CPT_)      // 64 threads per phase-1 block (2 waves)
#define CH_   8                // rows per load chunk (8 x b128 per thread)

__device__ __forceinline__ float wave_max32(float x) {
#pragma unroll
  for (int off = 16; off > 0; off >>= 1)
    x = fmaxf(x, __shfl_xor(x, off, 32));
  return x;
}

__device__ __forceinline__ float wave_min32(float x) {
#pragma unroll
  for (int off = 16; off > 0; off >>= 1)
    x = fminf(x, __shfl_xor(x, off, 32));
  return x;
}

// ---------------------------------------------------------------------------
// Phase 1: thread (b, seg, c4) streams 256 rows of a float4 (4 channels) with
// ping-pong double buffering; keeps 4x top-8 and 4x bot-8 lists in registers.
// Consecutive lanes read consecutive float4s -> 512 B contiguous per wave
// load (global_load_b128), 8 loads in flight while the previous chunk is
// processed.
// ---------------------------------------------------------------------------
#define LOAD8(DST, R0)                                                        \
  _Pragma("unroll")                                                           \
  for (int j = 0; j < CH_; ++j) (DST)[j] = p[(size_t)((R0) + j) * stride4];   \
  __builtin_prefetch(p + (size_t)((R0) + 4 * CH_) * stride4, 0, 0);

#define PROC8(SRC)                                                            \
  _Pragma("unroll")                                                           \
  for (int j = 0; j < CH_; ++j) {                                             \
    const float4 v = (SRC)[j];                                                \
    const float xs[4] = {v.x, v.y, v.z, v.w};                                 \
    _Pragma("unroll")                                                         \
    for (int ch = 0; ch < CPT_; ++ch) {                                       \
      const float x = xs[ch];                                                 \
      if (x > t[ch][K_ - 1]) {                                                \
        t[ch][K_ - 1] = x;                                                    \
        _Pragma("unroll")                                                     \
        for (int i = K_ - 1; i > 0; --i) {                                    \
          const float hi = fmaxf(t[ch][i - 1], t[ch][i]);                     \
          const float lo = fminf(t[ch][i - 1], t[ch][i]);                     \
          t[ch][i - 1] = hi; t[ch][i] = lo;                                   \
        }                                                                     \
      }                                                                       \
      if (x < u[ch][K_ - 1]) {                                                \
        u[ch][K_ - 1] = x;                                                    \
        _Pragma("unroll")                                                     \
        for (int i = K_ - 1; i > 0; --i) {                                    \
          const float lo = fminf(u[ch][i - 1], u[ch][i]);                     \
          const float hi = fmaxf(u[ch][i - 1], u[ch][i]);                     \
          u[ch][i - 1] = lo; u[ch][i] = hi;                                   \
        }                                                                     \
      }                                                                       \
    }                                                                         \
  }

__global__ __launch_bounds__(TPB1_) void topk_phase1(const float4* __restrict__ in,
                                                     float* __restrict__ tops,
                                                     float* __restrict__ bots) {
  const int b   = blockIdx.x / S_;
  const int seg = blockIdx.x % S_;
  const int c4  = threadIdx.x;                 // float4 column (channel group)
  const int stride4 = C_ / 4;                  // 64 float4 per spatial row

  float t[CPT_][K_];                           // descending per channel
  float u[CPT_][K_];                           // ascending per channel
#pragma unroll
  for (int ch = 0; ch < CPT_; ++ch)
#pragma unroll
    for (int i = 0; i < K_; ++i) { t[ch][i] = -INFINITY; u[ch][i] = INFINITY; }

  const float4* p = in + ((size_t)b * HW_ + (size_t)seg * ROWS_) * stride4 + c4;

  float4 A[CH_], Bv[CH_];
  LOAD8(A, 0);
  int it = 0;
  for (; it < ROWS_ - 2 * CH_; it += 2 * CH_) {
    LOAD8(Bv, it + CH_);       // next chunk in flight while we process A
    PROC8(A);
    LOAD8(A, it + 2 * CH_);    // chunk after next in flight while we process B
    PROC8(Bv);
  }
  // it == ROWS_ - 2*CH_: two chunks left (rows it.. and it+CH_..)
  LOAD8(Bv, it + CH_);
  PROC8(A);
  PROC8(Bv);

#pragma unroll
  for (int ch = 0; ch < CPT_; ++ch) {
    const int c = c4 * CPT_ + ch;
    const size_t pb = (((size_t)b * C_ + c) * S_ + seg) * K_;
#pragma unroll
    for (int i = 0; i < K_; ++i) tops[pb + i] = t[ch][i];
#pragma unroll
    for (int i = 0; i < K_; ++i) bots[pb + i] = u[ch][i];
  }
}

// ---------------------------------------------------------------------------
// Phase 2: one wave32 per (b,c). Lane l loads segments 2l and 2l+1 (16
// contiguous floats, 4x b128), register-merges them into one 8-list, then
// 8x (wave reduce + ballot-pick-owner + register pop) extracts the global
// top-8 / bot-8 in order. Lane 0 writes the 16 outputs.
// ---------------------------------------------------------------------------
__global__ __launch_bounds__(256) void topk_phase2(const float* __restrict__ tops,
                                                   const float* __restrict__ bots,
                                                   float* __restrict__ out) {
  const int gt   = blockIdx.x * blockDim.x + threadIdx.x;
  const int wid  = gt >> 5;            // == b*C + c
  const int lane = threadIdx.x & 31;

  const size_t pb = ((size_t)wid * S_ + 2 * (size_t)lane) * K_;

  // ---- tops: merge two descending 8-lists, then wave-extract max 8x ----
  float a[K_], bb[K_], m[K_];
  {
    const float4 q0 = *(const float4*)(tops + pb);
    const float4 q1 = *(const float4*)(tops + pb + 4);
    const float4 q2 = *(const float4*)(tops + pb + 8);
    const float4 q3 = *(const float4*)(tops + pb + 12);
    a[0]=q0.x; a[1]=q0.y; a[2]=q0.z; a[3]=q0.w;
    a[4]=q1.x; a[5]=q1.y; a[6]=q1.z; a[7]=q1.w;
    bb[0]=q2.x; bb[1]=q2.y; bb[2]=q2.z; bb[3]=q2.w;
    bb[4]=q3.x; bb[5]=q3.y; bb[6]=q3.z; bb[7]=q3.w;
  }
#pragma unroll
  for (int k = 0; k < K_; ++k) {
    const bool ta = (a[0] >= bb[0]);
    m[k] = ta ? a[0] : bb[0];
#pragma unroll
    for (int i = 0; i < K_ - 1; ++i) {
      a[i]  = ta ? a[i + 1] : a[i];
      bb[i] = ta ? bb[i]    : bb[i + 1];
    }
    a[K_ - 1]  = ta ? -INFINITY : a[K_ - 1];
    bb[K_ - 1] = ta ? bb[K_ - 1] : -INFINITY;
  }

  float rt[K_];
#pragma unroll
  for (int k = 0; k < K_; ++k) {
    const float mx = wave_max32(m[0]);
    const unsigned long long bal = __ballot(m[0] == mx);  // wave32: low 32 bits
    const int owner = __ffsll(bal) - 1;
    if (lane == owner) {
#pragma unroll
      for (int i = 0; i < K_ - 1; ++i) m[i] = m[i + 1];
      m[K_ - 1] = -INFINITY;
    }
    rt[k] = mx;
  }

  // ---- bots: merge two ascending 8-lists, then wave-extract min 8x ----
  {
    const float4 q0 = *(const float4*)(bots + pb);
    const float4 q1 = *(const float4*)(bots + pb + 4);
    const float4 q2 = *(const float4*)(bots + pb + 8);
    const float4 q3 = *(const float4*)(bots + pb + 12);
    a[0]=q0.x; a[1]=q0.y; a[2]=q0.z; a[3]=q0.w;
    a[4]=q1.x; a[5]=q1.y; a[6]=q1.z; a[7]=q1.w;
    bb[0]=q2.x; bb[1]=q2.y; bb[2]=q2.z; bb[3]=q2.w;
    bb[4]=q3.x; bb[5]=q3.y; bb[6]=q3.z; bb[7]=q3.w;
  }
#pragma unroll
  for (int k = 0; k < K_; ++k) {
    const bool ta = (a[0] <= bb[0]);
    m[k] = ta ? a[0] : bb[0];
#pragma unroll
    for (int i = 0; i < K_ - 1; ++i) {
      a[i]  = ta ? a[i + 1] : a[i];
      bb[i] = ta ? bb[i]    : bb[i + 1];
    }
    a[K_ - 1]  = ta ? INFINITY : a[K_ - 1];
    bb[K_ - 1] = ta ? bb[K_ - 1] : INFINITY;
  }

  float rb[K_];
#pragma unroll
  for (int k = 0; k < K_; ++k) {
    const float mn = wave_min32(m[0]);
    const unsigned long long bal = __ballot(m[0] == mn);
    const int owner = __ffsll(bal) - 1;
    if (lane == owner) {
#pragma unroll
      for (int i = 0; i < K_ - 1; ++i) m[i] = m[i + 1];
      m[K_ - 1] = INFINITY;
    }
    rb[k] = mn;
  }

  if (lane == 0) {
    const size_t ob = (size_t)wid * (2 * K_);
#pragma unroll
    for (int k = 0; k < K_; ++k) out[ob + k] = rt[k];
#pragma unroll
    for (int k = 0; k < K_; ++k) out[ob + K_ + k] = rb[k];
  }
}

extern "C" void kernel_launch(void* const* d_in, const int* in_sizes, int n_in,
                              void* d_out, int out_size, void* d_ws, size_t ws_size,
                              hipStream_t stream) {
  (void)in_sizes; (void)n_in; (void)out_size; (void)ws_size;

  const float4* in  = (const float4*)d_in[0];
  float*        out = (float*)d_out;

  // Scratch: tops[B][C][S][8] and bots[B][C][S][8] -> 2 * 8 MiB = 16 MiB.
  float* tops = (float*)d_ws;
  float* bots = tops + (size_t)B_ * C_ * S_ * K_;

  // Phase 1: B*S = 1024 blocks x 64 threads (2 waves) = 2048 waves, each with
  // 8 x b128 loads (4 KB/wave) in flight while processing -> ~8 MB MLP.
  topk_phase1<<<dim3(B_ * S_), dim3(TPB1_), 0, stream>>>(in, tops, bots);

  // Phase 2: one wave32 per (b,c): 4096 waves -> 512 blocks x 256 threads.
  topk_phase2<<<dim3((B_ * C_ * 32) / 256), dim3(256), 0, stream>>>(tops, bots, out);
}